// PNL_68040871903658
// MI455X (gfx1250) — compile-verified
//
#include <hip/hip_runtime.h>
#include <math.h>

typedef float v2f __attribute__((ext_vector_type(2)));
typedef float v8f __attribute__((ext_vector_type(8)));

// ---------------------------------------------------------------------------
// Problem constants (from reference): B=4, high=low=64, inner=32, H=W=96
// ---------------------------------------------------------------------------
#define BB     4
#define NPIX   9216      // 96*96
#define NL2    4608      // folded columns
#define CNT_BN 36864     // B*H*W per-channel count for BN

// workspace layout (in floats)
#define OFF_TH   ((size_t)0)                    // theta proj  B*32*9216
#define OFF_G    ((size_t)1179648)              // g proj
#define OFF_PHI  ((size_t)2359296)              // phi proj
#define OFF_Y    ((size_t)3538944)              // y (B*64*4608)
#define OFF_WY   ((size_t)4718592)              // w_y (B*64*9216)
#define OFF_KV   ((size_t)7077888)              // kv  (B*64*64)
#define OFF_ST   ((size_t)7094272)              // stats: sum[64], sumsq[64]
#define OFF_SS   ((size_t)7094400)              // scale[64], shift[64]

#define WMMA_F32(a, x, c) \
    __builtin_amdgcn_wmma_f32_16x16x4_f32(false, (a), false, (x), (short)0, (c), false, false)

// ---------------------------------------------------------------------------
// Wave-level fp32 WMMA GEMM, register-blocked 4x over N:
//   Out[b] = scale * (A[b] @ X[b]) + bias
//   A element (m,k) at A[m*sam + k*sak]  (strided -> handles kv^T case)
//   X is K x N row-major (batch stride K*N); Out is M x N (batch stride M*N)
//   Each wave computes a 16(M) x 64(N) strip: one A fragment feeds 4 WMMAs
//   per K-step (K is compile-time -> fully unrolled). DO_STATS fuses the
//   per-channel sum/sumsq reduction for BatchNorm (shuffles + atomics).
// ---------------------------------------------------------------------------
template <int K, bool DO_STATS>
__global__ void gemm_wmma_f32(const float* __restrict__ A, long strideAb,
                              int sam, int sak,
                              const float* __restrict__ X,
                              const float* __restrict__ bias,
                              float* __restrict__ Out,
                              int M, int N, float scale,
                              float* __restrict__ stats)
{
    const int warp = threadIdx.x >> 5;
    const int lane = threadIdx.x & 31;
    const int half = lane >> 4;       // 0: lanes 0-15, 1: lanes 16-31
    const int l16  = lane & 15;
    const int koff = half << 1;       // K offset within group of 4

    const int nt0   = (blockIdx.x * (blockDim.x >> 5) + warp) << 2; // 4 tiles
    const int mbase = blockIdx.y << 4;
    const int col   = (nt0 << 4) + l16;
    const int b     = blockIdx.z;

    A   += (size_t)b * (size_t)strideAb;
    X   += (size_t)b * (size_t)K * (size_t)N + col;
    Out += (size_t)b * (size_t)M * (size_t)N + col;

    const float* Arow = A + (size_t)(mbase + l16) * (size_t)sam;

    v8f acc0 = {0.f,0.f,0.f,0.f,0.f,0.f,0.f,0.f};
    v8f acc1 = acc0, acc2 = acc0, acc3 = acc0;

#pragma unroll
    for (int k = 0; k < K; k += 4) {
        // A 16x4 tile: lane(0-15)=row M, halves hold K={0,1} / {2,3}
        v2f a;
        a.x = Arow[(size_t)(k + koff)     * (size_t)sak];
        a.y = Arow[(size_t)(k + koff + 1) * (size_t)sak];
        // B 4x16 tiles: lane(&15)=col N, halves hold K={0,1} / {2,3}
        const float* Xr0 = X + (size_t)(k + koff)     * (size_t)N;
        const float* Xr1 = X + (size_t)(k + koff + 1) * (size_t)N;
        v2f x0, x1, x2, x3;
        x0.x = Xr0[0];  x0.y = Xr1[0];
        x1.x = Xr0[16]; x1.y = Xr1[16];
        x2.x = Xr0[32]; x2.y = Xr1[32];
        x3.x = Xr0[48]; x3.y = Xr1[48];
        acc0 = WMMA_F32(a, x0, acc0);
        acc1 = WMMA_F32(a, x1, acc1);
        acc2 = WMMA_F32(a, x2, acc2);
        acc3 = WMMA_F32(a, x3, acc3);
    }

#pragma unroll
    for (int r = 0; r < 8; ++r) {
        // C/D layout: VGPR r holds row M = r (lanes 0-15) / r+8 (lanes 16-31)
        const int row = mbase + r + (half << 3);
        const float bv = bias ? bias[row] : 0.f;
        const float v0 = acc0[r] * scale + bv;
        const float v1 = acc1[r] * scale + bv;
        const float v2 = acc2[r] * scale + bv;
        const float v3 = acc3[r] * scale + bv;
        float* o = Out + (size_t)row * (size_t)N;
        o[0] = v0; o[16] = v1; o[32] = v2; o[48] = v3;

        if (DO_STATS) {
            float s1 = v0 + v1 + v2 + v3;
            float s2 = v0 * v0 + v1 * v1 + v2 * v2 + v3 * v3;
            // reduce across the 16 lanes of this half-wave (masks < 16)
#pragma unroll
            for (int m = 8; m >= 1; m >>= 1) {
                s1 += __shfl_xor(s1, m, 32);
                s2 += __shfl_xor(s2, m, 32);
            }
            if (l16 == 0) {
                atomicAdd(&stats[row],     s1);
                atomicAdd(&stats[M + row], s2);
            }
        }
    }
}

// ---------------------------------------------------------------------------
// kv[b] (64x64) += phi_fold(64 x 4608) @ g_fold^T  — K split across blocks,
// accumulated with f32 atomics (kv zero-initialized beforehand).
// ---------------------------------------------------------------------------
__global__ void kv_wmma_f32(const float* __restrict__ phi,
                            const float* __restrict__ g,
                            float* __restrict__ kv,
                            int Kl, int KC)
{
    const int lane = threadIdx.x & 31;
    const int half = lane >> 4;
    const int l16  = lane & 15;
    const int koff = half << 1;

    const int kc    = blockIdx.x;
    const int tile  = blockIdx.y;        // 0..15 -> 4x4 tiles of 64x64
    const int b     = blockIdx.z;
    const int mbase = (tile >> 2) << 4;
    const int nbase = (tile & 3) << 4;

    phi += (size_t)b * 64 * (size_t)Kl;
    g   += (size_t)b * 64 * (size_t)Kl;
    kv  += (size_t)b * 64 * 64;

    const float* prow = phi + (size_t)(mbase + l16) * (size_t)Kl; // A row (M=c)
    const float* grow = g   + (size_t)(nbase + l16) * (size_t)Kl; // B col (N=l)

    const int k0 = kc * KC;
    v8f acc = {0.f,0.f,0.f,0.f,0.f,0.f,0.f,0.f};
#pragma unroll 4
    for (int k = k0; k < k0 + KC; k += 4) {
        v2f a, x;
        a.x = prow[k + koff];
        a.y = prow[k + koff + 1];
        x.x = grow[k + koff];        // B(k,n) = g[n*Kl + k]  (B^T access)
        x.y = grow[k + koff + 1];
        acc = WMMA_F32(a, x, acc);
    }

    const int col = nbase + l16;
#pragma unroll
    for (int r = 0; r < 8; ++r) {
        const int row = mbase + r + (half << 3);
        atomicAdd(&kv[row * 64 + col], acc[r]);
    }
}

// ---------------------------------------------------------------------------
// helpers
// ---------------------------------------------------------------------------
__global__ void zero_f32(float* __restrict__ p, int n)
{
    int i = blockIdx.x * blockDim.x + threadIdx.x;
    if (i < n) p[i] = 0.f;
}

__global__ void bn_prep(const float* __restrict__ stats,
                        const float* __restrict__ gamma,
                        const float* __restrict__ beta,
                        float* __restrict__ ss, int M, float invCount)
{
    int c = threadIdx.x;
    if (c < M) {
        float mean = stats[c] * invCount;
        float var  = stats[M + c] * invCount - mean * mean;
        float inv  = rsqrtf(var + 1e-5f);
        float sc   = gamma[c] * inv;
        ss[c]      = sc;
        ss[M + c]  = beta[c] - mean * sc;
    }
}

__global__ void bn_finalize(const float* __restrict__ wy,
                            const float* __restrict__ xh,
                            const float* __restrict__ ss,
                            float* __restrict__ out, int total4)
{
    int i = blockIdx.x * blockDim.x + threadIdx.x;
    if (i >= total4) return;
    const int c = ((i * 4) / NPIX) & 63;          // channel (constant over 4)
    const float sc = ss[c], sh = ss[64 + c];
    const float4 w = ((const float4*)wy)[i];
    const float4 x = ((const float4*)xh)[i];
    float4 o;
    o.x = sc * w.x + sh + x.x;
    o.y = sc * w.y + sh + x.y;
    o.z = sc * w.z + sh + x.z;
    o.w = sc * w.w + sh + x.w;
    ((float4*)out)[i] = o;
}

// ---------------------------------------------------------------------------
extern "C" void kernel_launch(void* const* d_in, const int* in_sizes, int n_in,
                              void* d_out, int out_size, void* d_ws, size_t ws_size,
                              hipStream_t stream)
{
    const float* x_h     = (const float*)d_in[0];
    const float* x_l     = (const float*)d_in[1];
    const float* g_w     = (const float*)d_in[2];
    const float* g_b     = (const float*)d_in[3];
    const float* theta_w = (const float*)d_in[4];
    const float* theta_b = (const float*)d_in[5];
    const float* phi_w   = (const float*)d_in[6];
    const float* phi_b   = (const float*)d_in[7];
    const float* w_w     = (const float*)d_in[8];
    const float* w_b     = (const float*)d_in[9];
    const float* gamma   = (const float*)d_in[10];
    const float* beta    = (const float*)d_in[11];

    float* ws  = (float*)d_ws;
    float* out = (float*)d_out;

    // zero kv + stats (harness poisons ws; must be deterministic every call)
    zero_f32<<<dim3(65), dim3(256), 0, stream>>>(ws + OFF_KV, 16384 + 128);

    // projections: (32 x 64) @ (64 x 9216) + bias, per batch.
    // block = 4 waves, each wave = one 16x64 strip; 576 n-tiles / 16 = 36
    gemm_wmma_f32<64, false><<<dim3(36, 2, BB), dim3(128), 0, stream>>>(
        theta_w, 0, 64, 1, x_h, theta_b, ws + OFF_TH, 32, NPIX, 1.f, nullptr);
    gemm_wmma_f32<64, false><<<dim3(36, 2, BB), dim3(128), 0, stream>>>(
        g_w, 0, 64, 1, x_l, g_b, ws + OFF_G, 32, NPIX, 1.f, nullptr);
    gemm_wmma_f32<64, false><<<dim3(36, 2, BB), dim3(128), 0, stream>>>(
        phi_w, 0, 64, 1, x_l, phi_b, ws + OFF_PHI, 32, NPIX, 1.f, nullptr);

    // kv (64x64 per batch) = phi_fold @ g_fold^T, K=4608 split into 9 chunks
    kv_wmma_f32<<<dim3(9, 16, BB), dim3(32), 0, stream>>>(
        ws + OFF_PHI, ws + OFF_G, ws + OFF_KV, NL2, 512);

    // yT (64 x 4608 per batch) = kv^T @ theta_fold, scaled by 1/Nl2
    // A = kv^T -> strides sam=1, sak=64; per-batch A stride 64*64
    gemm_wmma_f32<64, false><<<dim3(18, 4, BB), dim3(128), 0, stream>>>(
        ws + OFF_KV, 4096, 1, 64, ws + OFF_TH, nullptr, ws + OFF_Y,
        64, NL2, 1.f / (float)NL2, nullptr);

    // w_y (64 x 9216 per batch) = w_w(64x32) @ y(32x9216) + w_b,
    // fused per-channel sum / sumsq accumulation for BatchNorm
    gemm_wmma_f32<32, true><<<dim3(36, 4, BB), dim3(128), 0, stream>>>(
        w_w, 0, 32, 1, ws + OFF_Y, w_b, ws + OFF_WY, 64, NPIX, 1.f,
        ws + OFF_ST);

    // fold BN stats into per-channel scale/shift
    bn_prep<<<dim3(1), dim3(64), 0, stream>>>(
        ws + OFF_ST, gamma, beta, ws + OFF_SS, 64, 1.f / (float)CNT_BN);

    // out = scale[c]*w_y + shift[c] + x_h   (float4 vectorized)
    const int total4 = (BB * 64 * NPIX) / 4;   // 589824
    bn_finalize<<<dim3(total4 / 256), dim3(256), 0, stream>>>(
        ws + OFF_WY, x_h, ws + OFF_SS, out, total4);
}